// MSELoss_29446295781453
// MI455X (gfx1250) — compile-verified
//
#include <hip/hip_runtime.h>
#include <hip/hip_bf16.h>

typedef __attribute__((ext_vector_type(2))) float v2f;
typedef __attribute__((ext_vector_type(8))) float v8f;

#define BB      16
#define HH      640
#define WW      640
#define HW      (HH * WW)
#define NPIX    (BB * HW)
#define TOPK    512
#define BORDER  8
#define GK      15
#define GR      7
#define CAND_MAX 24576
#define TILES_X 40           // 640/16
#define TILES_PER_IMG (TILES_X * TILES_X)
#define TOTAL_TILES (BB * TILES_PER_IMG)

// ---------------------------------------------------------------- init
__global__ void initKernel(float* normAcc, float* lossAcc, int* c1, int* c2) {
    int t = threadIdx.x;
    if (t == 0) { *normAcc = 0.f; *lossAcc = 0.f; }
    if (t < BB) { c1[t] = 0; c2[t] = 0; }
}

// ------------------------------------------------- warp + mask + norm
__global__ __launch_bounds__(256) void warpKernel(
    const float* __restrict__ score2, const float* __restrict__ homo,
    float* __restrict__ wout, unsigned char* __restrict__ mask,
    float* __restrict__ normAcc)
{
    int gid = blockIdx.x * blockDim.x + threadIdx.x;
    float mval = 0.f;
    if (gid < NPIX) {
        int b = gid / HW, p = gid % HW, y = p / WW, x = p % WW;
        const float* Hm = homo + b * 9;
        float fx = (float)x, fy = (float)y;
        float X = Hm[0]*fx + Hm[1]*fy + Hm[2];
        float Y = Hm[3]*fx + Hm[4]*fy + Hm[5];
        float Z = Hm[6]*fx + Hm[7]*fy + Hm[8];
        if (fabsf(Z) < 1e-8f) Z = 1e-8f;
        float sx = X / Z, sy = Y / Z;
        float x0f = floorf(sx), y0f = floorf(sy);
        float x1f = x0f + 1.f, y1f = y0f + 1.f;
        float ax = sx - x0f, ay = sy - y0f;
        float wa = (1.f-ax)*(1.f-ay), wb = (1.f-ax)*ay, wc = ax*(1.f-ay), wd = ax*ay;
        const float* src = score2 + (size_t)b * HW;
        auto samp = [&](float yy, float xx, float& flag) -> float {
            bool valid = (xx >= 0.f) && (xx <= (float)(WW-1)) &&
                         (yy >= 0.f) && (yy <= (float)(HH-1));
            flag = valid ? 1.f : 0.f;
            float xc = fminf(fmaxf(xx, 0.f), (float)(WW-1));
            float yc = fminf(fmaxf(yy, 0.f), (float)(HH-1));
            int xi = (int)xc, yi = (int)yc;
            return valid ? src[yi*WW + xi] : 0.f;
        };
        float va, vb, vc, vd;
        float ga = samp(y0f, x0f, va);
        float gb = samp(y1f, x0f, vb);
        float gc = samp(y0f, x1f, vc);
        float gd = samp(y1f, x1f, vd);
        float wv   = wa*ga + wb*gb + wc*gc + wd*gd;
        float ones = wa*va + wb*vb + wc*vc + wd*vd;
        mval = (ones > 0.f) ? 1.f : 0.f;
        bool inB = (y >= BORDER) && (y < HH-BORDER) && (x >= BORDER) && (x < WW-BORDER);
        wout[gid] = inB ? wv : 0.f;
        mask[gid] = (unsigned char)mval;
    }
    // norm reduction
    for (int off = 16; off > 0; off >>= 1) mval += __shfl_down(mval, off, 32);
    __shared__ float red[8];
    int lane = threadIdx.x & 31, wv2 = threadIdx.x >> 5;
    if (lane == 0) red[wv2] = mval;
    __syncthreads();
    if (threadIdx.x == 0) {
        float s = 0.f;
        for (int i = 0; i < (int)(blockDim.x >> 5); ++i) s += red[i];
        atomicAdd(normAcc, s);
    }
}

// ------------------------------------------------ 5x5 NMS + candidates
__global__ __launch_bounds__(256) void nmsCandKernel(
    const float* __restrict__ s, int applyBorder,
    float* __restrict__ cval, int* __restrict__ cidx, int* __restrict__ ctr)
{
    int gid = blockIdx.x * blockDim.x + threadIdx.x;
    if (gid >= NPIX) return;
    int b = gid / HW, p = gid % HW, y = p / WW, x = p % WW;
    const float* src = s + (size_t)b * HW;
    auto val = [&](int yy, int xx) -> float {
        if (yy < 0 || yy >= HH || xx < 0 || xx >= WW) return -INFINITY;
        float v = src[yy*WW + xx];
        if (applyBorder && (yy < BORDER || yy >= HH-BORDER || xx < BORDER || xx >= WW-BORDER))
            v = 0.f;
        return v;
    };
    float c = val(y, x);
    if (c <= 0.f) return;                 // NMS_THRESH == 0
    float mx = -INFINITY;
    for (int dy = -2; dy <= 2; ++dy)
        for (int dx = -2; dx <= 2; ++dx)
            mx = fmaxf(mx, val(y+dy, x+dx));
    if (c == mx) {
        int pos = atomicAdd(&ctr[b], 1);
        if (pos < CAND_MAX) {
            cval[b*CAND_MAX + pos] = c;
            cidx[b*CAND_MAX + pos] = y*WW + x;
        }
    }
}

// ----------------------------------- per-image top-512 (radix + bitonic)
__global__ __launch_bounds__(512) void topkKernel(
    const float* __restrict__ cval, const int* __restrict__ cidx,
    const int* __restrict__ ctr,
    float* __restrict__ kvout, int* __restrict__ kiout,
    float* __restrict__ kpOut, int writeKp)
{
    int b = blockIdx.x;
    int tid = threadIdx.x;
    int N = ctr[b]; if (N > CAND_MAX) N = CAND_MAX;
    const float* vv = cval + b*CAND_MAX;
    const int*   ii = cidx + b*CAND_MAX;

    __shared__ unsigned hist[256];
    __shared__ int selB1, selB2;
    __shared__ unsigned selAll, needIn, cnt;
    __shared__ unsigned long long keys[2048];

    for (int i = tid; i < 256; i += 512) hist[i] = 0u;
    if (tid == 0) { cnt = 0u; selAll = 0u; selB2 = 0; }
    __syncthreads();
    for (int i = tid; i < N; i += 512) {
        unsigned bt = __float_as_uint(vv[i]);
        atomicAdd(&hist[(bt >> 23) & 0xFF], 1u);
    }
    __syncthreads();
    if (tid == 0) {
        unsigned cum = 0; int sel = -1; unsigned need = TOPK;
        for (int h = 255; h >= 0; --h) {
            if (cum + hist[h] >= TOPK) { sel = h; need = TOPK - cum; break; }
            cum += hist[h];
        }
        selB1 = sel; needIn = need;
        if (sel < 0) selAll = 1u;
    }
    __syncthreads();
    int b1 = selB1;
    if (!selAll) {
        for (int i = tid; i < 256; i += 512) hist[i] = 0u;
        __syncthreads();
        for (int i = tid; i < N; i += 512) {
            unsigned bt = __float_as_uint(vv[i]);
            if ((int)((bt >> 23) & 0xFF) == b1)
                atomicAdd(&hist[(bt >> 15) & 0xFF], 1u);
        }
        __syncthreads();
        if (tid == 0) {
            unsigned cum2 = 0; int sel2 = 0;
            for (int h = 255; h >= 0; --h) {
                if (cum2 + hist[h] >= needIn) { sel2 = h; break; }
                cum2 += hist[h];
            }
            selB2 = sel2;
        }
        __syncthreads();
    }
    unsigned thr16 = selAll ? 0u : (((unsigned)b1 << 8) | (unsigned)selB2);
    for (int i = tid; i < N; i += 512) {
        unsigned bt = __float_as_uint(vv[i]);
        if (selAll || (bt >> 15) >= thr16) {
            unsigned pos = atomicAdd(&cnt, 1u);
            if (pos < 2048u)
                keys[pos] = ((unsigned long long)bt << 32) | (unsigned)(~(unsigned)ii[i]);
        }
    }
    __syncthreads();
    unsigned M2 = cnt < 2048u ? cnt : 2048u;
    for (unsigned i = tid; i < 2048u; i += 512u) if (i >= M2) keys[i] = 0ULL;
    __syncthreads();
    // bitonic sort, descending
    for (unsigned k = 2; k <= 2048; k <<= 1) {
        for (unsigned j = k >> 1; j > 0; j >>= 1) {
            for (unsigned i = tid; i < 2048u; i += 512u) {
                unsigned p = i ^ j;
                if (p > i) {
                    bool up = ((i & k) == 0u);
                    unsigned long long a = keys[i], q = keys[p];
                    bool sw = up ? (a < q) : (a > q);
                    if (sw) { keys[i] = q; keys[p] = a; }
                }
            }
            __syncthreads();
        }
    }
    for (int j = tid; j < TOPK; j += 512) {
        unsigned long long key = keys[j];
        float v = __uint_as_float((unsigned)(key >> 32));
        int idx = (int)(~(unsigned)(key & 0xFFFFFFFFu));
        if (key == 0ULL) { v = 0.f; idx = 0; }
        kvout[b*TOPK + j] = v;
        kiout[b*TOPK + j] = idx;
        if (writeKp) {
            kpOut[(size_t)(b*TOPK + j)*2 + 0] = (float)(idx / WW);
            kpOut[(size_t)(b*TOPK + j)*2 + 1] = (float)(idx % WW);
        }
    }
}

// ------------------------------------------------ zero + scatter
__global__ void zero2Kernel(float* a, float* b) {
    int g = blockIdx.x * blockDim.x + threadIdx.x;
    if (g < NPIX) { a[g] = 0.f; b[g] = 0.f; }
}

__global__ void scatterKernel(const float* kv1, const int* ki1, float* m1,
                              const float* kv2, const int* ki2, float* m2)
{
    int g = blockIdx.x * blockDim.x + threadIdx.x;
    if (g >= BB * TOPK) return;
    int b = g / TOPK;
    float v1 = kv1[g]; if (v1 > 0.f) m1[(size_t)b*HW + ki1[g]] = v1;
    float v2 = kv2[g]; if (v2 > 0.f) m2[(size_t)b*HW + ki2[g]] = v2;
}

// ---------------- fused WMMA separable 15x15 Gaussian + masked MSE ----
// out16x16 = Wcol[16x32] x ( Patch[32x32] x Wrow[32x16] ), band Gaussian.
// Patch staging uses CDNA5 async global->LDS loads (ASYNCcnt).
__global__ __launch_bounds__(128) void convLossKernel(
    const float* __restrict__ s1, const float* __restrict__ s2,
    const unsigned char* __restrict__ mask, float* __restrict__ lossAcc)
{
    __shared__ float gwz[48];          // gwz[d+16] = g[d], 0<=d<15 else 0
    __shared__ float patch[4][32*32];
    __shared__ float mid[4][32*16];
    __shared__ float redbuf[4];

    int tid = threadIdx.x;
    if (tid < 48) {
        int d = tid - 16;
        float v = 0.f;
        if (d >= 0 && d < GK) {
            float s = 0.f;
            for (int k = 0; k < GK; ++k) {
                float t = (float)(k - GR);
                s += expf(-t*t*2.0f);          // 1/(2*sigma^2) = 2, sigma = 0.5
            }
            float t = (float)(d - GR);
            v = expf(-t*t*2.0f) / s;
        }
        gwz[tid] = v;
    }
    __syncthreads();

    int wave = tid >> 5, lane = tid & 31;
    int lane15 = lane & 15, half = lane >> 4;
    int tileId = blockIdx.x * 4 + wave;
    int b = tileId / TILES_PER_IMG;
    int t = tileId % TILES_PER_IMG;
    int ty = (t / TILES_X) * 16, tx = (t % TILES_X) * 16;

    const float* maps[2] = { s1, s2 };
    float* P = patch[wave];
    float* M = mid[wave];
    v8f out[2];

    // prefetch the second map's tile while we chew on the first
    __builtin_prefetch(s2 + (size_t)b*HW + (size_t)ty*WW + tx, 0, 1);

    unsigned pbase = (unsigned)(size_t)(void*)P;    // LDS byte offset of this wave's patch

    for (int m = 0; m < 2; ++m) {
        // 1) zero only the pad region (rows/cols >= 30 or out of image) with DS stores
        for (int i = lane; i < 1024; i += 32) {
            int r = i >> 5, c = i & 31;
            int gy = ty - GR + r, gx = tx - GR + c;
            bool valid = (r < 30) && (c < 30) && (gy >= 0) && (gy < HH) &&
                         (gx >= 0) && (gx < WW);
            if (!valid) P[i] = 0.f;     // disjoint from async-written addresses
        }
        // 2) async global->LDS for the valid 30x30 region (GVS mode: SGPR base + VGPR off)
        for (int i = lane; i < 900; i += 32) {
            int r = i / 30, c = i - r * 30;
            int gy = ty - GR + r, gx = tx - GR + c;
            if (gy >= 0 && gy < HH && gx >= 0 && gx < WW) {
                unsigned ldsOff = pbase + (unsigned)((r * 32 + c) * 4);
                unsigned gOff = (unsigned)(((size_t)b * HW + (size_t)gy * WW + gx) * 4);
                asm volatile("global_load_async_to_lds_b32 %0, %1, %2"
                             :: "v"(ldsOff), "v"(gOff), "s"(maps[m])
                             : "memory");
            }
        }
        asm volatile("s_wait_asynccnt 0" ::: "memory");
        __syncthreads();

        // row pass: mid[32x16] = Patch[32x32] x Wrow[32x16], Wrow[j,n]=g[j-n]
        v8f acc0 = {}; v8f acc1 = {};
        for (int kc = 0; kc < 8; ++kc) {
            int k0 = kc*4 + half*2;
            v2f a0, a1, bb;
            a0.x = P[lane15*32 + k0];        a0.y = P[lane15*32 + k0 + 1];
            a1.x = P[(16+lane15)*32 + k0];   a1.y = P[(16+lane15)*32 + k0 + 1];
            bb.x = gwz[k0     - lane15 + 16];
            bb.y = gwz[k0 + 1 - lane15 + 16];
            acc0 = __builtin_amdgcn_wmma_f32_16x16x4_f32(false, a0, false, bb,
                                                         (short)0, acc0, false, false);
            acc1 = __builtin_amdgcn_wmma_f32_16x16x4_f32(false, a1, false, bb,
                                                         (short)0, acc1, false, false);
        }
        for (int r = 0; r < 8; ++r) {
            M[(half*8 + r)*16 + lane15]      = acc0[r];
            M[(16 + half*8 + r)*16 + lane15] = acc1[r];
        }
        __syncthreads();

        // col pass: out[16x16] = Wcol[16x32] x mid[32x16], Wcol[m,j]=g[j-m]
        v8f o = {};
        for (int kc = 0; kc < 8; ++kc) {
            int k0 = kc*4 + half*2;
            v2f a, bb;
            a.x = gwz[k0     - lane15 + 16];
            a.y = gwz[k0 + 1 - lane15 + 16];
            bb.x = M[k0*16 + lane15];
            bb.y = M[(k0+1)*16 + lane15];
            o = __builtin_amdgcn_wmma_f32_16x16x4_f32(false, a, false, bb,
                                                      (short)0, o, false, false);
        }
        out[m] = o;
        __syncthreads();
    }

    // masked squared difference for this tile
    float part = 0.f;
    const unsigned char* mrow = mask + (size_t)b * HW;
    for (int r = 0; r < 8; ++r) {
        int gy = ty + half*8 + r;
        int gx = tx + lane15;
        float mk = (float)mrow[(size_t)gy*WW + gx];
        float d = out[0][r] - out[1][r];
        part += d * d * mk;
    }
    for (int off = 16; off > 0; off >>= 1) part += __shfl_down(part, off, 32);
    if (lane == 0) redbuf[wave] = part;
    __syncthreads();
    if (tid == 0) {
        float s = redbuf[0] + redbuf[1] + redbuf[2] + redbuf[3];
        atomicAdd(lossAcc, s);
    }
}

// ---------------------------------------------------------------- final
__global__ void finalKernel(const float* lossAcc, const float* normAcc, float* outp) {
    float n = *normAcc;
    outp[0] = (n > 0.f ? (*lossAcc) / n : 0.f) * 1.0f;   // LOSS_LAMBDA = 1
}

// ================================================================ launch
extern "C" void kernel_launch(void* const* d_in, const int* in_sizes, int n_in,
                              void* d_out, int out_size, void* d_ws, size_t ws_size,
                              hipStream_t stream)
{
    const float* score1 = (const float*)d_in[0];
    const float* score2 = (const float*)d_in[1];
    const float* homo   = (const float*)d_in[2];
    float* outp = (float*)d_out;

    // workspace bump allocator (256B aligned)
    size_t off = 0;
    auto alloc = [&](size_t bytes) -> void* {
        void* r = (void*)((char*)d_ws + off);
        off = (off + bytes + 255) & ~(size_t)255;
        return r;
    };
    float* buf0 = (float*)alloc((size_t)NPIX * 4);          // warped score2 -> g2 sparse map
    float* buf1 = (float*)alloc((size_t)NPIX * 4);          // s1 sparse map
    unsigned char* mask = (unsigned char*)alloc((size_t)NPIX);
    float* cand_val1 = (float*)alloc((size_t)BB * CAND_MAX * 4);
    int*   cand_idx1 = (int*)  alloc((size_t)BB * CAND_MAX * 4);
    float* cand_val2 = (float*)alloc((size_t)BB * CAND_MAX * 4);
    int*   cand_idx2 = (int*)  alloc((size_t)BB * CAND_MAX * 4);
    float* kv1 = (float*)alloc((size_t)BB * TOPK * 4);
    int*   ki1 = (int*)  alloc((size_t)BB * TOPK * 4);
    float* kv2 = (float*)alloc((size_t)BB * TOPK * 4);
    int*   ki2 = (int*)  alloc((size_t)BB * TOPK * 4);
    float* normAcc = (float*)alloc(4);
    float* lossAcc = (float*)alloc(4);
    int*   ctr1 = (int*)alloc(BB * 4);
    int*   ctr2 = (int*)alloc(BB * 4);

    const int TPB = 256;
    const int pixBlocks = (NPIX + TPB - 1) / TPB;

    initKernel<<<1, 64, 0, stream>>>(normAcc, lossAcc, ctr1, ctr2);
    warpKernel<<<pixBlocks, TPB, 0, stream>>>(score2, homo, buf0, mask, normAcc);
    nmsCandKernel<<<pixBlocks, TPB, 0, stream>>>(score1, 1, cand_val1, cand_idx1, ctr1);
    topkKernel<<<BB, 512, 0, stream>>>(cand_val1, cand_idx1, ctr1, kv1, ki1, outp + 1, 1);
    nmsCandKernel<<<pixBlocks, TPB, 0, stream>>>(buf0, 0, cand_val2, cand_idx2, ctr2);
    topkKernel<<<BB, 512, 0, stream>>>(cand_val2, cand_idx2, ctr2, kv2, ki2, nullptr, 0);
    zero2Kernel<<<pixBlocks, TPB, 0, stream>>>(buf0, buf1);
    scatterKernel<<<(BB*TOPK + TPB - 1)/TPB, TPB, 0, stream>>>(kv1, ki1, buf1, kv2, ki2, buf0);
    convLossKernel<<<TOTAL_TILES / 4, 128, 0, stream>>>(buf1, buf0, mask, lossAcc);
    finalKernel<<<1, 1, 0, stream>>>(lossAcc, normAcc, outp);
}